// INNV4E4A6_37417755083508
// MI455X (gfx1250) — compile-verified
//
#include <hip/hip_runtime.h>
#include <math.h>

#define NPTS 400
#define HIDN 64
#define ZMINF (-3.0f)
#define ZMAXF (3.0f)
#define NB 1024
#define NT 256

typedef __attribute__((ext_vector_type(2))) float v2f;
typedef __attribute__((ext_vector_type(8))) float v8f;

// ---------------------------------------------------------------------------
// Workspace layout (bytes):
//   0          : float F[400]
//   1600       : float w[400]
//   3200       : float scale, shift
//   4096       : double bsum[NB]
//   4096+8*NB  : double bsq [NB]
// ---------------------------------------------------------------------------

// ---------------- Kernel 1: grid MLP + cumsum (uses WMMA f32 16x16x4) -------
__global__ __launch_bounds__(256) void grid_mlp_kernel(
    const float* __restrict__ W1, const float* __restrict__ b1,
    const float* __restrict__ W2, const float* __restrict__ b2,
    const float* __restrict__ W3, const float* __restrict__ b3,
    float* __restrict__ Fout, float* __restrict__ wout)
{
    __shared__ float W2T[HIDN * HIDN];   // W2T[j][i] = W2[i][j]
    __shared__ float gbuf[NPTS];
    __shared__ float wbuf[NPTS];
    __shared__ float sc[512];

    const int   tid = threadIdx.x;
    const float dt  = (ZMAXF - ZMINF) / (float)(NPTS - 1);

    for (int i = tid; i < HIDN * HIDN; i += NT) {
        int r = i >> 6, c = i & 63;
        W2T[c * HIDN + r] = W2[i];
    }
    __syncthreads();

    const int wave = tid >> 5;
    const int lane = tid & 31;
    const int half = lane >> 4;       // 0: lanes 0-15, 1: lanes 16-31
    const int n    = lane & 15;       // N/M index within half

    // 25 row tiles of 16 rows each (400 = 25*16), 8 waves
    for (int tile = wave; tile < NPTS / 16; tile += 8) {
        const int   row = tile * 16 + n;              // A-matrix row (M)
        const float t   = ZMINF + (float)row * dt;

        // Layer 1: h1 values this lane needs for its A fragments.
        // f32 16x4 A layout: lanes 0-15 hold K={0,1}, lanes 16-31 hold K={2,3}.
        float aa[16], ab[16];
        #pragma unroll
        for (int k = 0; k < 16; ++k) {
            int j = 4 * k + 2 * half;
            aa[k] = tanhf(t * W1[j]     + b1[j]);
            ab[k] = tanhf(t * W1[j + 1] + b1[j + 1]);
        }

        float gsum[8];
        #pragma unroll
        for (int v = 0; v < 8; ++v) gsum[v] = 0.0f;

        for (int ct = 0; ct < 4; ++ct) {             // 4 column tiles of 16
            const int   col  = ct * 16 + n;          // output neuron (N)
            const float bias = b2[col];
            v8f acc;
            #pragma unroll
            for (int v = 0; v < 8; ++v) acc[v] = bias;

            #pragma unroll
            for (int k = 0; k < 16; ++k) {           // K = 64 in chunks of 4
                const int j = 4 * k + 2 * half;
                v2f a; a.x = aa[k];                 a.y = ab[k];
                v2f b; b.x = W2T[j * HIDN + col];   b.y = W2T[(j + 1) * HIDN + col];
                acc = __builtin_amdgcn_wmma_f32_16x16x4_f32(
                        false, a, false, b, (short)0, acc, false, false);
            }

            // Layer 3 partial: g[p] += tanh(h2[p][col]) * W3[col]
            const float w3c = W3[col];
            #pragma unroll
            for (int v = 0; v < 8; ++v) {
                float contrib = tanhf(acc[v]) * w3c;
                #pragma unroll
                for (int off = 1; off < 16; off <<= 1)
                    contrib += __shfl_xor(contrib, off, 32);  // stays in half
                gsum[v] += contrib;
            }
        }
        if (n == 0) {
            const int   base = tile * 16 + 8 * half;  // D rows: v + 8*half
            const float bb   = b3[0];
            #pragma unroll
            for (int v = 0; v < 8; ++v) gbuf[base + v] = gsum[v] + bb;
        }
    }
    __syncthreads();

    // w = exp(clip(g, -1, 1))
    for (int i = tid; i < NPTS; i += NT)
        wbuf[i] = expf(fminf(fmaxf(gbuf[i], -1.0f), 1.0f));
    __syncthreads();

    // Trapezoid increments then inclusive Hillis-Steele scan over 512 slots.
    {
        const int i0 = tid, i1 = tid + NT;
        sc[i0] = (i0 < NPTS - 1) ? 0.5f * (wbuf[i0] + wbuf[i0 + 1]) * dt : 0.0f;
        sc[i1] = (i1 < NPTS - 1) ? 0.5f * (wbuf[i1] + wbuf[i1 + 1]) * dt : 0.0f;
        __syncthreads();
        for (int off = 1; off < 512; off <<= 1) {
            float v0 = sc[i0] + ((i0 >= off) ? sc[i0 - off] : 0.0f);
            float v1 = sc[i1] + ((i1 >= off) ? sc[i1 - off] : 0.0f);
            __syncthreads();
            sc[i0] = v0; sc[i1] = v1;
            __syncthreads();
        }
    }
    for (int i = tid; i < NPTS; i += NT) {
        Fout[i] = (i == 0) ? 0.0f : sc[i - 1];
        wout[i] = wbuf[i];
    }
}

// ---------------- Per-sample interpolation ----------------------------------
__device__ __forceinline__ float evalF(float zz, const float* __restrict__ Fs,
                                       const float* __restrict__ Ws, float dt)
{
    float pos  = (zz - ZMINF) / dt;
    int   idx  = (int)floorf(pos);
    idx        = idx < 0 ? 0 : (idx > NPTS - 2 ? NPTS - 2 : idx);
    float frac = fminf(fmaxf(pos - (float)idx, 0.0f), 1.0f);
    float r    = Fs[idx] + frac * Ws[idx] * dt;
    if (zz < ZMINF) r = (zz - ZMINF) * Ws[0];                       // F[0] == 0
    if (zz > ZMAXF) r = Fs[NPTS - 1] + (zz - ZMAXF) * Ws[NPTS - 1];
    return r;
}

// ---------------- Kernel 2: stats pass (no atomics, deterministic) ----------
__global__ __launch_bounds__(256) void pass1_kernel(
    const float* __restrict__ z, int B,
    const float* __restrict__ Ftab, const float* __restrict__ wtab,
    double* __restrict__ bsum, double* __restrict__ bsq)
{
    __shared__ float  Fs[NPTS], Ws[NPTS];
    __shared__ double rs[NT], rq[NT];
    const int tid = threadIdx.x;
    for (int i = tid; i < NPTS; i += NT) { Fs[i] = Ftab[i]; Ws[i] = wtab[i]; }
    __syncthreads();

    const float dt = (ZMAXF - ZMINF) / (float)(NPTS - 1);
    float s = 0.0f, s2 = 0.0f;
    const int     nvec = B >> 2;
    const float4* z4   = (const float4*)z;
    for (int i = blockIdx.x * NT + tid; i < nvec; i += NB * NT) {
        float4 v = z4[i];
        float f;
        f = evalF(v.x, Fs, Ws, dt); s += f; s2 += f * f;
        f = evalF(v.y, Fs, Ws, dt); s += f; s2 += f * f;
        f = evalF(v.z, Fs, Ws, dt); s += f; s2 += f * f;
        f = evalF(v.w, Fs, Ws, dt); s += f; s2 += f * f;
    }
    for (int i = (nvec << 2) + blockIdx.x * NT + tid; i < B; i += NB * NT) {
        float f = evalF(z[i], Fs, Ws, dt); s += f; s2 += f * f;
    }
    rs[tid] = (double)s; rq[tid] = (double)s2;
    __syncthreads();
    for (int off = NT / 2; off > 0; off >>= 1) {
        if (tid < off) { rs[tid] += rs[tid + off]; rq[tid] += rq[tid + off]; }
        __syncthreads();
    }
    if (tid == 0) { bsum[blockIdx.x] = rs[0]; bsq[blockIdx.x] = rq[0]; }
}

// ---------------- Kernel 3: finalize stats ----------------------------------
__global__ __launch_bounds__(256) void finalize_kernel(
    const double* __restrict__ bsum, const double* __restrict__ bsq,
    const float* __restrict__ a_raw, const float* __restrict__ b_param,
    int B, float* __restrict__ scale_shift)
{
    __shared__ double rs[NT], rq[NT];
    const int tid = threadIdx.x;
    double s = 0.0, q = 0.0;
    for (int i = tid; i < NB; i += NT) { s += bsum[i]; q += bsq[i]; }
    rs[tid] = s; rq[tid] = q;
    __syncthreads();
    for (int off = NT / 2; off > 0; off >>= 1) {
        if (tid < off) { rs[tid] += rs[tid + off]; rq[tid] += rq[tid + off]; }
        __syncthreads();
    }
    if (tid == 0) {
        const double S  = rs[0], Q = rq[0];
        const double mu = S / (double)B;
        double var = (Q - S * S / (double)B) / (double)(B - 1);
        if (var < 0.0) var = 0.0;
        float sig = fmaxf((float)sqrt(var), 1e-6f);
        float ar  = a_raw[0];
        float a   = (ar > 20.0f) ? ar : log1pf(expf(ar));   // softplus
        a += 1e-3f;
        const float scale = a / sig;
        scale_shift[0] = scale;
        scale_shift[1] = b_param[0] - scale * (float)mu;
    }
}

// ---------------- Kernel 4: output pass -------------------------------------
__global__ __launch_bounds__(256) void pass2_kernel(
    const float* __restrict__ z, int B,
    const float* __restrict__ Ftab, const float* __restrict__ wtab,
    const float* __restrict__ scale_shift, float* __restrict__ out)
{
    __shared__ float Fs[NPTS], Ws[NPTS];
    const int tid = threadIdx.x;
    for (int i = tid; i < NPTS; i += NT) { Fs[i] = Ftab[i]; Ws[i] = wtab[i]; }
    __syncthreads();

    const float dt    = (ZMAXF - ZMINF) / (float)(NPTS - 1);
    const float scale = scale_shift[0];
    const float shift = scale_shift[1];

    const int     nvec = B >> 2;
    const float4* z4   = (const float4*)z;
    float4*       o4   = (float4*)out;
    for (int i = blockIdx.x * NT + tid; i < nvec; i += NB * NT) {
        float4 v = z4[i], r;
        r.x = scale * evalF(v.x, Fs, Ws, dt) + shift;
        r.y = scale * evalF(v.y, Fs, Ws, dt) + shift;
        r.z = scale * evalF(v.z, Fs, Ws, dt) + shift;
        r.w = scale * evalF(v.w, Fs, Ws, dt) + shift;
        o4[i] = r;
    }
    for (int i = (nvec << 2) + blockIdx.x * NT + tid; i < B; i += NB * NT)
        out[i] = scale * evalF(z[i], Fs, Ws, dt) + shift;
}

// ---------------------------------------------------------------------------
extern "C" void kernel_launch(void* const* d_in, const int* in_sizes, int n_in,
                              void* d_out, int out_size, void* d_ws, size_t ws_size,
                              hipStream_t stream)
{
    const float* z       = (const float*)d_in[0];
    const float* W1      = (const float*)d_in[1];
    const float* b1      = (const float*)d_in[2];
    const float* W2      = (const float*)d_in[3];
    const float* b2      = (const float*)d_in[4];
    const float* W3      = (const float*)d_in[5];
    const float* b3      = (const float*)d_in[6];
    const float* a_raw   = (const float*)d_in[7];
    const float* b_param = (const float*)d_in[8];
    const int B = in_sizes[0];

    char*   ws    = (char*)d_ws;
    float*  Ftab  = (float*)(ws + 0);
    float*  wtab  = (float*)(ws + 1600);
    float*  sstab = (float*)(ws + 3200);
    double* bsum  = (double*)(ws + 4096);
    double* bsq   = (double*)(ws + 4096 + 8 * NB);
    float*  out   = (float*)d_out;

    grid_mlp_kernel<<<1, NT, 0, stream>>>(W1, b1, W2, b2, W3, b3, Ftab, wtab);
    pass1_kernel<<<NB, NT, 0, stream>>>(z, B, Ftab, wtab, bsum, bsq);
    finalize_kernel<<<1, NT, 0, stream>>>(bsum, bsq, a_raw, b_param, B, sstab);
    pass2_kernel<<<NB, NT, 0, stream>>>(z, B, Ftab, wtab, sstab, out);
}